// ADMBlock_3685081940036
// MI455X (gfx1250) — compile-verified
//
#include <hip/hip_runtime.h>
#include <math.h>

#define N_RES   16384
#define KNBR    32
#define DMODEL  256
#define PPAIR   64
#define NHEAD   8
#define HDIM    32
#define NBATCH  16
#define NCHAIN  64
#define RELPOSN 65
#define RBFBINS 16

typedef float v2f __attribute__((ext_vector_type(2)));
typedef float v8f __attribute__((ext_vector_type(8)));

__device__ __forceinline__ float gelu_f(float x) {
  float x3 = x * x * x;
  return 0.5f * x * (1.0f + tanhf(0.7978845608028654f * (x + 0.044715f * x3)));
}

// ---------------------------------------------------------------------------
// FP32 WMMA GEMM: C[rows x Dout] = X[rows x Din] @ W[Din x Dout] (+bias)(gelu)
// grid.x = rows/16, grid.y = Dout/128, block = 256 (8 waves, one 16x16 tile
// each). Uses V_WMMA_F32_16X16X4_F32 with the CDNA5 ISA operand layouts:
//   A 16x4: lanes 0-15 hold K={k0,k0+1} for M=lane; lanes 16-31 K={k0+2,k0+3}
//   B 4x16: vgpr0 = rows K={k0,k0+2} striped over lanes, vgpr1 = K={k0+1,k0+3}
//   C/D  : vgpr j -> row j + 8*(lane>=16), col = lane&15
// ---------------------------------------------------------------------------
__global__ void gemm_wmma_f32(const float* __restrict__ X, const float* __restrict__ W,
                              const float* __restrict__ bias, float* __restrict__ C,
                              int Din, int Dout, int apply_gelu) {
  const int wave = threadIdx.x >> 5;
  const int lane = threadIdx.x & 31;
  const int row0 = blockIdx.x << 4;
  const int col0 = ((blockIdx.y << 3) + wave) << 4;
  const int m    = lane & 15;
  const int half = lane >> 4;
  v8f acc = {0.f, 0.f, 0.f, 0.f, 0.f, 0.f, 0.f, 0.f};
  const float* Xrow = X + (size_t)(row0 + m) * Din;
  const float* Wc   = W + col0 + m;
  for (int k0 = 0; k0 < Din; k0 += 4) {
    const int ka = k0 + 2 * half;
    v2f a;
    a.x = Xrow[ka];
    a.y = Xrow[ka + 1];
    v2f b;
    b.x = Wc[(size_t)ka * Dout];
    b.y = Wc[(size_t)(ka + 1) * Dout];
    acc = __builtin_amdgcn_wmma_f32_16x16x4_f32(false, a, false, b, (short)0, acc,
                                                false, false);
  }
  const float bv = bias ? bias[col0 + m] : 0.0f;
#pragma unroll
  for (int j = 0; j < 8; ++j) {
    const int r = row0 + j + 8 * half;
    float out = acc[j] + bv;
    if (apply_gelu) out = gelu_f(out);
    C[(size_t)r * Dout + col0 + m] = out;
  }
}

__global__ void zero_f32(float* p, int n) {
  int i = blockIdx.x * blockDim.x + threadIdx.x;
  if (i < n) p[i] = 0.0f;
}

__global__ void seg_accum(const float* __restrict__ h, const int* __restrict__ bi,
                          const int* __restrict__ ci, const unsigned char* __restrict__ mask,
                          float* __restrict__ segB, float* __restrict__ segC,
                          float* __restrict__ cntB, float* __restrict__ cntC) {
  const int n = blockIdx.x;
  const int d = threadIdx.x;  // 512
  const float w  = mask[n] ? 1.0f : 0.0f;
  const float hv = h[(size_t)n * 512 + d] * w;
  atomicAdd(&segB[(size_t)bi[n] * 512 + d], hv);
  atomicAdd(&segC[(size_t)ci[n] * 512 + d], hv);
  if (d == 0) { atomicAdd(&cntB[bi[n]], w); atomicAdd(&cntC[ci[n]], w); }
}

__global__ void seg_finalize(float* segB, float* segC, const float* cntB, const float* cntC) {
  const int idx = blockIdx.x * blockDim.x + threadIdx.x;
  if (idx < NBATCH * 512) {
    const int row = idx / 512;
    segB[idx] = gelu_f(segB[idx] / fmaxf(cntB[row], 1e-6f));
  } else if (idx < (NBATCH + NCHAIN) * 512) {
    const int i2 = idx - NBATCH * 512;
    const int row = i2 / 512;
    segC[i2] = gelu_f(segC[i2] / fmaxf(cntC[row], 1e-6f));
  }
}

__global__ void resi1_kernel(const float* __restrict__ local, const float* __restrict__ incr,
                             const float* __restrict__ PB, const float* __restrict__ PC,
                             const int* __restrict__ bi, const int* __restrict__ ci,
                             const float* __restrict__ lnsc, const float* __restrict__ lnbi,
                             float* __restrict__ local_out, float* __restrict__ inc_out) {
  const int n = blockIdx.x, d = threadIdx.x;
  const float gu = PB[(size_t)bi[n] * DMODEL + d] + PC[(size_t)ci[n] * DMODEL + d];
  const float x = local[(size_t)n * DMODEL + d] + gu;
  __shared__ float s1[8], s2[8];
  float sum = x, sq = x * x;
  for (int o = 16; o > 0; o >>= 1) { sum += __shfl_xor(sum, o, 32); sq += __shfl_xor(sq, o, 32); }
  if ((threadIdx.x & 31) == 0) { s1[threadIdx.x >> 5] = sum; s2[threadIdx.x >> 5] = sq; }
  __syncthreads();
  if (threadIdx.x == 0) {
    float a = 0, b = 0;
    for (int i = 0; i < 8; ++i) { a += s1[i]; b += s2[i]; }
    s1[0] = a; s2[0] = b;
  }
  __syncthreads();
  const float mean = s1[0] * (1.0f / DMODEL);
  const float var  = s2[0] * (1.0f / DMODEL) - mean * mean;
  local_out[(size_t)n * DMODEL + d] = (x - mean) * rsqrtf(var + 1e-5f) * lnsc[d] + lnbi[d];
  inc_out[(size_t)n * DMODEL + d]   = incr[(size_t)n * DMODEL + d] + gu;
}

__global__ void resi_kernel(const float* __restrict__ x_in, const float* __restrict__ add,
                            const float* __restrict__ lnsc, const float* __restrict__ lnbi,
                            float* __restrict__ x_out, float* __restrict__ inc_out) {
  const int n = blockIdx.x, d = threadIdx.x;
  const size_t idx = (size_t)n * DMODEL + d;
  const float a = add[idx];
  const float x = x_in[idx] + a;
  __shared__ float s1[8], s2[8];
  float sum = x, sq = x * x;
  for (int o = 16; o > 0; o >>= 1) { sum += __shfl_xor(sum, o, 32); sq += __shfl_xor(sq, o, 32); }
  if ((threadIdx.x & 31) == 0) { s1[threadIdx.x >> 5] = sum; s2[threadIdx.x >> 5] = sq; }
  __syncthreads();
  if (threadIdx.x == 0) {
    float aa = 0, bb = 0;
    for (int i = 0; i < 8; ++i) { aa += s1[i]; bb += s2[i]; }
    s1[0] = aa; s2[0] = bb;
  }
  __syncthreads();
  const float mean = s1[0] * (1.0f / DMODEL);
  const float var  = s2[0] * (1.0f / DMODEL) - mean * mean;
  x_out[idx] = (x - mean) * rsqrtf(var + 1e-5f) * lnsc[d] + lnbi[d];
  inc_out[idx] += a;
}

__global__ void combine_small(const float* Wrp, const float* Wdist, const float* Wrot,
                              const float* Wvec, const float* Wb,
                              float* Wrpb, float* Wdb, float* Wrotb, float* Wvb) {
  const int total = (RELPOSN + RBFBINS + 9 + 15) * NHEAD;
  for (int i = threadIdx.x; i < total; i += blockDim.x) {
    const int r = i / NHEAD, h = i % NHEAD;
    const float* src; float* dst; int rr;
    if (r < RELPOSN)                    { src = Wrp;   dst = Wrpb;  rr = r; }
    else if (r < RELPOSN + RBFBINS)     { src = Wdist; dst = Wdb;   rr = r - RELPOSN; }
    else if (r < RELPOSN + RBFBINS + 9) { src = Wrot;  dst = Wrotb; rr = r - RELPOSN - RBFBINS; }
    else                                { src = Wvec;  dst = Wvb;   rr = r - RELPOSN - RBFBINS - 9; }
    float s = 0.0f;
    for (int p = 0; p < PPAIR; ++p) s += src[rr * PPAIR + p] * Wb[p * NHEAD + h];
    dst[rr * NHEAD + h] = s;
  }
}

__device__ __forceinline__ void compute_frame(const float* __restrict__ pos, int idx,
                                              float e[3][3], float ca[3]) {
  const float* pr = pos + (size_t)idx * 15;
  const float nv0 = pr[0], nv1 = pr[1], nv2 = pr[2];
  ca[0] = pr[3]; ca[1] = pr[4]; ca[2] = pr[5];
  float c0 = pr[6] - ca[0], c1 = pr[7] - ca[1], c2 = pr[8] - ca[2];
  float inv = 1.0f / (sqrtf(c0 * c0 + c1 * c1 + c2 * c2) + 1e-8f);
  e[0][0] = c0 * inv; e[0][1] = c1 * inv; e[0][2] = c2 * inv;
  const float u0 = nv0 - ca[0], u1 = nv1 - ca[1], u2 = nv2 - ca[2];
  const float dd = u0 * e[0][0] + u1 * e[0][1] + u2 * e[0][2];
  float w0 = u0 - dd * e[0][0], w1 = u1 - dd * e[0][1], w2 = u2 - dd * e[0][2];
  inv = 1.0f / (sqrtf(w0 * w0 + w1 * w1 + w2 * w2) + 1e-8f);
  e[1][0] = w0 * inv; e[1][1] = w1 * inv; e[1][2] = w2 * inv;
  e[2][0] = e[0][1] * e[1][2] - e[0][2] * e[1][1];
  e[2][1] = e[0][2] * e[1][0] - e[0][0] * e[1][2];
  e[2][2] = e[0][0] * e[1][1] - e[0][1] * e[1][0];
}

__global__ void pair_bias_kernel(const float* __restrict__ pos, const int* __restrict__ ri,
                                 const int* __restrict__ ci, const int* __restrict__ nb,
                                 const float* __restrict__ Wrpb, const float* __restrict__ Wdb,
                                 const float* __restrict__ Wrotb, const float* __restrict__ Wvb,
                                 float* __restrict__ biasbuf) {
  const int n = blockIdx.x, k = threadIdx.x;  // block = 32
  const int j = nb[(size_t)n * KNBR + k];
  const bool valid = (j != -1);
  const int jj = (j < 0) ? 0 : j;
  float en[3][3], ej[3][3], can[3], caj[3];
  compute_frame(pos, n, en, can);
  compute_frame(pos, jj, ej, caj);
  float bh[NHEAD];
  int dd = ri[jj] - ri[n];
  dd = min(max(dd, -32), 32) + 32;
  const bool sc = (ci[jj] == ci[n]) && valid;
  for (int h = 0; h < NHEAD; ++h) bh[h] = sc ? Wrpb[dd * NHEAD + h] : 0.0f;
  const float dx = caj[0] - can[0], dy = caj[1] - can[1], dz = caj[2] - can[2];
  const float dist = sqrtf(dx * dx + dy * dy + dz * dz);
  const float sigma = 22.0f / (float)RBFBINS;
  const float inv2s2 = 1.0f / (2.0f * sigma * sigma);
  for (int b = 0; b < RBFBINS; ++b) {
    const float cb = 22.0f * (float)b / (float)(RBFBINS - 1);
    const float t = dist - cb;
    const float e = expf(-t * t * inv2s2);
    for (int h = 0; h < NHEAD; ++h) bh[h] += e * Wdb[b * NHEAD + h];
  }
  for (int a = 0; a < 3; ++a)
    for (int b = 0; b < 3; ++b) {
      const float r = en[a][0] * ej[b][0] + en[a][1] * ej[b][1] + en[a][2] * ej[b][2];
      const int i = a * 3 + b;
      for (int h = 0; h < NHEAD; ++h) bh[h] += r * Wrotb[i * NHEAD + h];
    }
  const float* pj = pos + (size_t)jj * 15;
  for (int p = 0; p < 5; ++p) {
    const float r0 = pj[p * 3 + 0] - can[0];
    const float r1 = pj[p * 3 + 1] - can[1];
    const float r2 = pj[p * 3 + 2] - can[2];
    for (int a = 0; a < 3; ++a) {
      const float v = 0.1f * (en[a][0] * r0 + en[a][1] * r1 + en[a][2] * r2);
      const int i = p * 3 + a;
      for (int h = 0; h < NHEAD; ++h) bh[h] += v * Wvb[i * NHEAD + h];
    }
  }
  float* out = biasbuf + ((size_t)n * KNBR + k) * NHEAD;
  for (int h = 0; h < NHEAD; ++h) out[h] = bh[h];
}

__global__ void attention_kernel(const float* __restrict__ q, const float* __restrict__ kk,
                                 const float* __restrict__ vv, const float* __restrict__ biasbuf,
                                 const int* __restrict__ nb, const unsigned char* __restrict__ mask,
                                 float* __restrict__ obuf) {
  const int n = blockIdx.x;
  const int h = threadIdx.x >> 5;
  const int lane = threadIdx.x & 31;
  __shared__ float qs[DMODEL];
  qs[threadIdx.x] = q[(size_t)n * DMODEL + threadIdx.x];
  __syncthreads();
  const int j = nb[(size_t)n * KNBR + lane];
  const bool valid = (j != -1);
  const int jj = (j < 0) ? 0 : j;
  const bool mOK = valid && (mask[n] != 0) && (mask[jj] != 0);
  const float* krow = kk + (size_t)jj * DMODEL + h * HDIM;
  float logit = 0.0f;
#pragma unroll
  for (int d = 0; d < HDIM; ++d) logit += qs[h * HDIM + d] * krow[d];
  logit *= 0.17677669529663687f;  // 1/sqrt(32)
  logit += biasbuf[((size_t)n * KNBR + lane) * NHEAD + h];
  if (!mOK) logit = -1e9f;
  float mx = logit;
  for (int o = 16; o > 0; o >>= 1) mx = fmaxf(mx, __shfl_xor(mx, o, 32));
  const float e = expf(logit - mx);
  float s = e;
  for (int o = 16; o > 0; o >>= 1) s += __shfl_xor(s, o, 32);
  const float a = e / s;
  float acc = 0.0f;
  for (int k2 = 0; k2 < KNBR; ++k2) {
    const float ak = __shfl(a, k2, 32);
    const int jk = __shfl(jj, k2, 32);
    acc += ak * vv[(size_t)jk * DMODEL + h * HDIM + lane];  // coalesced across lanes
  }
  obuf[(size_t)n * DMODEL + h * HDIM + lane] = acc;
}

extern "C" void kernel_launch(void* const* d_in, const int* in_sizes, int n_in,
                              void* d_out, int out_size, void* d_ws, size_t ws_size,
                              hipStream_t stream) {
  (void)in_sizes; (void)n_in; (void)out_size; (void)ws_size;
  const float* local = (const float*)d_in[0];
  const float* incr  = (const float*)d_in[1];
  const float* pos   = (const float*)d_in[2];
  const float* Wg1 = (const float*)d_in[3];  const float* bg1 = (const float*)d_in[4];
  const float* Wg2 = (const float*)d_in[5];  const float* bg2 = (const float*)d_in[6];
  const float* Wrp = (const float*)d_in[7];  const float* Wdist = (const float*)d_in[8];
  const float* Wrot = (const float*)d_in[9]; const float* Wvec = (const float*)d_in[10];
  const float* Wq = (const float*)d_in[11];  const float* Wk = (const float*)d_in[12];
  const float* Wv = (const float*)d_in[13];  const float* Wb = (const float*)d_in[14];
  const float* Wo = (const float*)d_in[15];  const float* bo = (const float*)d_in[16];
  const float* Wm1 = (const float*)d_in[17]; const float* bm1 = (const float*)d_in[18];
  const float* Wm2 = (const float*)d_in[19]; const float* bm2 = (const float*)d_in[20];
  const float* ln1s = (const float*)d_in[21]; const float* ln1b = (const float*)d_in[22];
  const float* ln2s = (const float*)d_in[23]; const float* ln2b = (const float*)d_in[24];
  const float* ln3s = (const float*)d_in[25]; const float* ln3b = (const float*)d_in[26];
  const int* ri = (const int*)d_in[27];
  const int* ci = (const int*)d_in[28];
  const int* bi = (const int*)d_in[29];
  const int* nb = (const int*)d_in[30];
  const unsigned char* mask = (const unsigned char*)d_in[31];

  float* ws = (float*)d_ws;
  size_t off = 0;
  auto alloc = [&](size_t nElem) { float* p = ws + off; off += nElem; return p; };
  float* h512  = alloc((size_t)N_RES * 512);
  float* bufL  = alloc((size_t)N_RES * DMODEL);
  float* bufT  = alloc((size_t)N_RES * DMODEL);
  float* qb    = alloc((size_t)N_RES * DMODEL);
  float* kb    = alloc((size_t)N_RES * DMODEL);
  float* vb    = alloc((size_t)N_RES * DMODEL);
  float* ob    = alloc((size_t)N_RES * DMODEL);
  float* biasb = alloc((size_t)N_RES * KNBR * NHEAD);
  float* segB  = alloc(NBATCH * 512);
  float* segC  = alloc(NCHAIN * 512);
  float* cntB  = alloc(NBATCH);
  float* cntC  = alloc(NCHAIN);
  float* PB    = alloc(NBATCH * DMODEL);
  float* PC    = alloc(NCHAIN * DMODEL);
  float* Wrpb  = alloc(RELPOSN * NHEAD);
  float* Wdb   = alloc(RBFBINS * NHEAD);
  float* Wrotb = alloc(9 * NHEAD);
  float* Wvb   = alloc(15 * NHEAD);

  float* outLocal = (float*)d_out;
  float* outInc   = outLocal + (size_t)N_RES * DMODEL;

  const int segZero = NBATCH * 512 + NCHAIN * 512 + NBATCH + NCHAIN;
  zero_f32<<<(segZero + 255) / 256, 256, 0, stream>>>(segB, segZero);

  gemm_wmma_f32<<<dim3(N_RES / 16, 512 / 128), 256, 0, stream>>>(local, Wg1, bg1, h512, DMODEL, 512, 0);
  seg_accum<<<N_RES, 512, 0, stream>>>(h512, bi, ci, mask, segB, segC, cntB, cntC);
  seg_finalize<<<((NBATCH + NCHAIN) * 512 + 255) / 256, 256, 0, stream>>>(segB, segC, cntB, cntC);
  gemm_wmma_f32<<<dim3(NBATCH / 16, 2), 256, 0, stream>>>(segB, Wg2, bg2, PB, 512, DMODEL, 0);
  gemm_wmma_f32<<<dim3(NCHAIN / 16, 2), 256, 0, stream>>>(segC, Wg2 + 512 * DMODEL, nullptr, PC, 512, DMODEL, 0);
  resi1_kernel<<<N_RES, DMODEL, 0, stream>>>(local, incr, PB, PC, bi, ci, ln1s, ln1b, bufL, outInc);

  combine_small<<<1, 256, 0, stream>>>(Wrp, Wdist, Wrot, Wvec, Wb, Wrpb, Wdb, Wrotb, Wvb);
  pair_bias_kernel<<<N_RES, KNBR, 0, stream>>>(pos, ri, ci, nb, Wrpb, Wdb, Wrotb, Wvb, biasb);

  gemm_wmma_f32<<<dim3(N_RES / 16, 2), 256, 0, stream>>>(bufL, Wq, nullptr, qb, DMODEL, DMODEL, 0);
  gemm_wmma_f32<<<dim3(N_RES / 16, 2), 256, 0, stream>>>(bufL, Wk, nullptr, kb, DMODEL, DMODEL, 0);
  gemm_wmma_f32<<<dim3(N_RES / 16, 2), 256, 0, stream>>>(bufL, Wv, nullptr, vb, DMODEL, DMODEL, 0);

  attention_kernel<<<N_RES, DMODEL, 0, stream>>>(qb, kb, vb, biasb, nb, mask, ob);

  gemm_wmma_f32<<<dim3(N_RES / 16, 2), 256, 0, stream>>>(ob, Wo, bo, bufT, DMODEL, DMODEL, 0);
  resi_kernel<<<N_RES, DMODEL, 0, stream>>>(bufL, bufT, ln2s, ln2b, bufL, outInc);

  gemm_wmma_f32<<<dim3(N_RES / 16, 512 / 128), 256, 0, stream>>>(bufL, Wm1, bm1, h512, DMODEL, 512, 1);
  gemm_wmma_f32<<<dim3(N_RES / 16, 2), 256, 0, stream>>>(h512, Wm2, bm2, bufT, 512, DMODEL, 0);
  resi_kernel<<<N_RES, DMODEL, 0, stream>>>(bufL, bufT, ln3s, ln3b, outLocal, outInc);
}